// RoIPool_12034498363587
// MI455X (gfx1250) — compile-verified
//
#include <hip/hip_runtime.h>
#include <stdint.h>

// Problem constants (match the reference).
#define C_    256
#define H_    56
#define W_    56
#define N_    128
#define OUT_  7
#define NEGV  (-3.4e38f)

// One block per channel. Stage the 56x56 plane (12,544 B) into LDS with the
// CDNA5 async global->LDS DMA engine (ASYNCcnt-tracked), overlap the proposal
// box computation with the DMA, then answer all 128 proposals x 49 bins for
// this channel out of LDS.
__global__ __launch_bounds__(256) void RoIPool_gfx1250_kernel(
    const float* __restrict__ fmap,       // [C][H][W]
    const float* __restrict__ proposals,  // [N][4]
    float* __restrict__ out)              // [N][C][7][7]
{
    __shared__ __align__(16) float sPlane[H_ * W_];  // 12,544 B
    __shared__ int4 sBox[N_];                        // x1, y1, rw, rh

    const int c = blockIdx.x;
    const int t = threadIdx.x;

    // ---- async DMA: this channel's plane -> LDS (global_load_async_to_lds) ----
    {
        const char* gBase   = (const char*)(fmap + (size_t)c * (H_ * W_));
        uint32_t    ldsBase = (uint32_t)(size_t)(&sPlane[0]);  // low 32 bits = LDS byte offset
        // 12,544 bytes = 784 x B128 transfers; ~3-4 per lane.
        for (int tr = t; tr < (H_ * W_ * 4) / 16; tr += 256) {
            uint32_t    ldsOff = ldsBase + tr * 16;
            const char* g      = gBase + tr * 16;
            asm volatile("global_load_async_to_lds_b128 %0, %1, off"
                         :: "v"(ldsOff), "v"(g)
                         : "memory");
        }
    }

    // ---- proposal boxes, computed while the DMA is in flight ----
    if (t < N_) {
        float4 p = ((const float4*)proposals)[t];
        int x1 = max((int)floorf(p.x * 0.0625f), 0);
        int y1 = max((int)floorf(p.y * 0.0625f), 0);
        int x2 = min((int)floorf(p.z * 0.0625f), W_);
        int y2 = min((int)floorf(p.w * 0.0625f), H_);
        sBox[t] = make_int4(x1, y1, max(x2 - x1, 0), max(y2 - y1, 0));
    }

    // Each wave drains its own async counter, then block barrier publishes LDS.
    asm volatile("s_wait_asynccnt 0" ::: "memory");
    __syncthreads();

    // ---- 7x7 adaptive max-pool per proposal, gathered from LDS ----
    for (int o = t; o < N_ * OUT_ * OUT_; o += 256) {
        int n = o / (OUT_ * OUT_);
        int r = o - n * (OUT_ * OUT_);
        int i = r / OUT_;        // h bin
        int j = r - i * OUT_;    // w bin
        int4 b = sBox[n];        // x1, y1, rw, rh

        // PyTorch adaptive bins: [i*L//7, ceil((i+1)*L/7))
        int hs = (i * b.w) / OUT_;
        int he = ((i + 1) * b.w + (OUT_ - 1)) / OUT_;
        int ws = (j * b.z) / OUT_;
        int we = ((j + 1) * b.z + (OUT_ - 1)) / OUT_;

        float m = NEGV;  // empty bins stay at NEG, matching the reference mask
        for (int yy = hs; yy < he; ++yy) {
            int y = min(max(b.y + yy, 0), H_ - 1);   // reference clips indices
            const float* row = &sPlane[y * W_];
            for (int xx = ws; xx < we; ++xx) {
                int x = min(max(b.x + xx, 0), W_ - 1);
                m = fmaxf(m, row[x]);
            }
        }
        out[((size_t)n * C_ + c) * (OUT_ * OUT_) + r] = m;
    }
}

extern "C" void kernel_launch(void* const* d_in, const int* in_sizes, int n_in,
                              void* d_out, int out_size, void* d_ws, size_t ws_size,
                              hipStream_t stream) {
    (void)in_sizes; (void)n_in; (void)out_size; (void)d_ws; (void)ws_size;
    const float* fmap      = (const float*)d_in[0];
    const float* proposals = (const float*)d_in[1];
    float*       out       = (float*)d_out;

    RoIPool_gfx1250_kernel<<<dim3(C_), dim3(256), 0, stream>>>(fmap, proposals, out);
}